// MPNNLayer_71700184039486
// MI455X (gfx1250) — compile-verified
//
#include <hip/hip_runtime.h>
#include <hip/hip_bf16.h>

// ext-vector types for WMMA fragments
typedef __attribute__((ext_vector_type(2))) float v2f;
typedef __attribute__((ext_vector_type(8))) float v8f;

#define D_FEAT 64   // D_IN == D_OUT == 64

// -----------------------------------------------------------------------------
// Kernel 1: out[n][j] = b[j]  (bias broadcast; runs before the scatter kernel on
// the same stream, so ordering is guaranteed)
// -----------------------------------------------------------------------------
__global__ void mpnn_bias_init(const float* __restrict__ bias,
                               float* __restrict__ out, int total) {
    int i = blockIdx.x * blockDim.x + threadIdx.x;
    if (i < total) out[i] = bias[i & (D_FEAT - 1)];
}

// -----------------------------------------------------------------------------
// Kernel 2: fused gather -> (16x64 msgs) @ W^T via V_WMMA_F32_16X16X4_F32
//           -> scatter-add rows into out[dst] with f32 HW atomics.
// One wave32 handles 16 edges. All 32 lanes stay active around every WMMA.
// -----------------------------------------------------------------------------
__global__ void mpnn_edge_wmma_scatter(const float* __restrict__ feat,
                                       const int*  __restrict__ src,
                                       const int*  __restrict__ dst,
                                       const float* __restrict__ W,   // [64][64] row-major (out,in)
                                       float* __restrict__ out,
                                       int n_edges) {
    const int lane      = threadIdx.x & 31;
    const int wave_blk  = threadIdx.x >> 5;
    const long long wid = (long long)blockIdx.x * (blockDim.x >> 5) + wave_blk;
    const int e0        = (int)(wid * 16);
    if (e0 >= n_edges) return;            // wave-uniform: EXEC stays all-1s

    const int half = lane >> 4;           // 0 -> lanes 0..15, 1 -> lanes 16..31
    const int l15  = lane & 15;
    const int koff = half * 2;            // K sub-offset carried by this lane half

    // --- Gather A fragments: row M = l15 of the 16x64 message tile ----------
    // A(16x4) f32 layout: lane(l15) holds [K=koff, K=koff+1] in 2 VGPRs.
    const int s = src[e0 + l15];
    const float* arow = feat + (long long)s * D_FEAT;
    v2f a[16];
    #pragma unroll
    for (int k = 0; k < 16; ++k) {
        a[k].x = arow[4 * k + koff];
        a[k].y = arow[4 * k + koff + 1];
    }

    // Destination node ids for the 8 output rows this lane's accumulator holds
    // (C/D layout: VGPR r holds row M = r + 8*half).
    int drow[8];
    #pragma unroll
    for (int r = 0; r < 8; ++r) drow[r] = dst[e0 + r + half * 8];

    // --- 4 column tiles of the 16x64 output ---------------------------------
    #pragma unroll
    for (int nt = 0; nt < 4; ++nt) {
        const int n = nt * 16 + l15;                  // output column for this lane
        const float* wrow = W + (long long)n * D_FEAT; // B[k][n] = W[n][k]
        v8f acc = {};                                  // bias already in d_out
        #pragma unroll
        for (int k = 0; k < 16; ++k) {
            v2f bb;
            bb.x = wrow[4 * k + koff];
            bb.y = wrow[4 * k + koff + 1];
            // D = A(16x4) x B(4x16) + C   -> v_wmma_f32_16x16x4_f32
            acc = __builtin_amdgcn_wmma_f32_16x16x4_f32(
                false, a[k], false, bb, (short)0, acc, false, false);
        }
        // Scatter: lane holds O[M = r + 8*half][N = n] in acc[r]
        #pragma unroll
        for (int r = 0; r < 8; ++r) {
            unsafeAtomicAdd(out + (long long)drow[r] * D_FEAT + n, acc[r]);
        }
    }
}

// -----------------------------------------------------------------------------
// Launch
// -----------------------------------------------------------------------------
extern "C" void kernel_launch(void* const* d_in, const int* in_sizes, int n_in,
                              void* d_out, int out_size, void* d_ws, size_t ws_size,
                              hipStream_t stream) {
    const float* feat = (const float*)d_in[0];   // [N, 64] f32
    const int*   src  = (const int*)d_in[1];     // [E]
    const int*   dst  = (const int*)d_in[2];     // [E]
    const float* W    = (const float*)d_in[3];   // [64, 64] f32 (out, in)
    const float* bias = (const float*)d_in[4];   // [64] f32
    float* out        = (float*)d_out;           // [N, 64] f32

    const int n_edges = in_sizes[1];
    const int total   = out_size;                // N * 64

    // 1) out = bias (broadcast)
    mpnn_bias_init<<<(total + 255) / 256, 256, 0, stream>>>(bias, out, total);

    // 2) fused gather + WMMA transform + atomic scatter
    const int waves  = (n_edges + 15) / 16;      // 16 edges per wave32
    const int wpb    = 256 / 32;                 // 8 waves per block
    const int blocks = (waves + wpb - 1) / wpb;
    mpnn_edge_wmma_scatter<<<blocks, 256, 0, stream>>>(feat, src, dst, W, out, n_edges);
}